// EQUIREC_ENC_33938831573236
// MI455X (gfx1250) — compile-verified
//
#include <hip/hip_runtime.h>

// Equirectangular multi-resolution grid encoding, 7 levels, F=2, N=2,097,152.
// Gather-bound kernel. wave32 layout, 28 independent global_load_b64 gathers
// issued before first use, LDS-staged transpose for coalesced output stores.
// Cache policy: levels 0-5 (89 MB total, high reuse) stay L2-resident (RT);
// level 6 (268 MB > 192 MB L2, reuse < 1) streamed non-temporal; input/output
// (touched once) non-temporal so they never evict the hot grids.

namespace {

constexpr int kNLev = 7;
constexpr int kRes[kNLev] = {64, 128, 256, 512, 1024, 2048, 4096};
constexpr int kF = 2;
constexpr int kOutF = kNLev * kF;        // 14 floats per point
constexpr int kBlock = 256;              // 8 wave32 waves
constexpr int kWaveSz = 32;
constexpr int kWavesPerBlock = kBlock / kWaveSz;

typedef float v2f __attribute__((ext_vector_type(2)));

struct Grids {
  const v2f* g[kNLev];                   // level grids, each (R+1, 2R+1, 2) f32
};

__global__ __launch_bounds__(kBlock) void equirec_enc_kernel(
    const v2f* __restrict__ x,           // [N] (lat, lon)
    Grids grids,
    float* __restrict__ out) {           // [N, 14]
  // Per-wave staging: 32 points * 14 floats; stride-14-word writes hit each
  // even LDS bank exactly once per wave (gcd(14,64)=2) -> conflict-free.
  __shared__ float stage[kWavesPerBlock][kWaveSz * kOutF];

  const int lane = threadIdx.x & (kWaveSz - 1);
  const int wave = threadIdx.x >> 5;
  const int tid  = blockIdx.x * kBlock + threadIdx.x;

  const v2f p = __builtin_nontemporal_load(&x[tid]);   // NT: read-once stream
  const float lat = p.x;
  const float lon = p.y;

  float wgt[kNLev][4];
  v2f   val[kNLev][4];

  // Phase 1+2: compute all addresses & weights, issue all 28 gathers before
  // any value is consumed -> max MLP; drained later by s_wait_loadcnt.
#pragma unroll
  for (int L = 0; L < kNLev; ++L) {
    const int R      = kRes[L];
    const int latDim = R + 1;
    const int lonDim = 2 * R + 1;
    // Match reference arithmetic: (lat+90)/180*R , lon/360*2*R
    // (divisions are hoisted by the compiler; done once, not per level)
    const float qlat = (lat + 90.0f) / 180.0f * (float)R;
    const float qlon = lon / 360.0f * (2.0f * (float)R);
    const int c0 = (int)floorf(qlat);
    const int c1 = (int)floorf(qlon);
    const v2f* __restrict__ g = grids.g[L];
#pragma unroll
    for (int c = 0; c < 4; ++c) {
      const int dx = c >> 1;             // DXDY = {0,0},{0,1},{1,0},{1,1}
      const int dy = c & 1;
      int i0 = c0 + dx;
      i0 = i0 < 0 ? 0 : (i0 > latDim - 1 ? latDim - 1 : i0);
      int i1 = c1 + dy;
      i1 = i1 < 0 ? 0 : (i1 > lonDim - 1 ? lonDim - 1 : i1);
      // Weights from CLAMPED coords (reference clamps before dist).
      wgt[L][c] = (1.0f - fabsf(qlat - (float)i0)) *
                  (1.0f - fabsf(qlon - (float)i1));
      // East-west wrap, then pole collapse (after weights).
      int j1 = (i1 == lonDim - 1) ? 0 : i1;
      j1 = (i0 == 0 || i0 == latDim - 1) ? 0 : j1;
      const int idx = i0 * lonDim + j1;
      if (L == kNLev - 1) {
        // 268 MB grid, line-reuse < 1: stream NT, don't evict hot levels.
        val[L][c] = __builtin_nontemporal_load(&g[idx]);
      } else {
        // 89 MB combined, high reuse: keep L2-resident (RT).
        val[L][c] = g[idx];              // per-lane global_load_b64 gather
      }
    }
  }

  // Phase 3: blend and stage per-wave into LDS (row-per-lane layout).
  float* ws = stage[wave];
#pragma unroll
  for (int L = 0; L < kNLev; ++L) {
    float ax = 0.0f, ay = 0.0f;
#pragma unroll
    for (int c = 0; c < 4; ++c) {
      ax = fmaf(wgt[L][c], val[L][c].x, ax);
      ay = fmaf(wgt[L][c], val[L][c].y, ay);
    }
    v2f r; r.x = ax; r.y = ay;
    *(v2f*)&ws[lane * kOutF + 2 * L] = r;              // merged ds stores
  }

  // Same-wave LDS ops are in-order on CDNA5; only block compiler reordering.
  __builtin_amdgcn_wave_barrier();

  // Coalesced output: this wave owns 448 contiguous floats of `out`.
  const long long waveBase =
      (long long)blockIdx.x * kBlock + (long long)wave * kWaveSz;
  v2f* o = (v2f*)(out + waveBase * kOutF);
  const v2f* s2 = (const v2f*)ws;
#pragma unroll
  for (int i = 0; i < kOutF / 2; ++i) {  // 7 x 256B-per-wave NT stores
    const v2f r = s2[i * kWaveSz + lane];              // ds_load, no conflicts
    __builtin_nontemporal_store(r, &o[i * kWaveSz + lane]);
  }
}

}  // namespace

extern "C" void kernel_launch(void* const* d_in, const int* in_sizes, int n_in,
                              void* d_out, int out_size, void* d_ws,
                              size_t ws_size, hipStream_t stream) {
  (void)n_in; (void)out_size; (void)d_ws; (void)ws_size;
  const v2f* x = (const v2f*)d_in[0];                  // [N,2] f32
  Grids grids;
  for (int i = 0; i < kNLev; ++i) {
    grids.g[i] = (const v2f*)d_in[1 + i];              // params[i]
  }
  float* out = (float*)d_out;                          // [N,14] f32

  const int n = in_sizes[0] / 2;                       // N_POINTS (2^21)
  const int blocks = n / kBlock;                       // exact: 8192
  equirec_enc_kernel<<<blocks, kBlock, 0, stream>>>(x, grids, out);
}